// shuffle_dim_no_darts_49340584297189
// MI455X (gfx1250) — compile-verified
//
#include <hip/hip_runtime.h>
#include <hip/hip_bf16.h>

typedef __attribute__((ext_vector_type(2))) float v2f;
typedef __attribute__((ext_vector_type(8))) float v8f;

#define B_ROWS 16384
#define N_FEAT 64
#define VOCAB  100000
#define EMB_D  16
#define FD     1024   // F*D
#define TEMP_  5.0f

// Per-column pseudo-random bijective batch permutation (B is a power of two,
// odd multiplier => bijection). Stands in for jax's threefry argsort shuffle.
__device__ __forceinline__ int perm_row(unsigned j, unsigned b) {
    const unsigned aj = (j * 2654435761u) | 1u;            // odd multiplier
    const unsigned cj = j * 0x9E3779B9u + 0x7F4A7C15u;     // offset
    return (int)((aj * b + cj) & (unsigned)(B_ROWS - 1));
}

__global__ __launch_bounds__(256) void gated_emb_dot_kernel(
    const int*   __restrict__ inp,     // [B, F]
    const float* __restrict__ emb,     // [F, V, D]
    const float* __restrict__ theta,   // [F*D]
    const float* __restrict__ weight,  // [F*D]
    float*       __restrict__ out)     // [B] (+1 for loss, written elsewhere)
{
    __shared__ float g_lds[FD];
    __shared__ float w_lds[FD];
    for (int j = threadIdx.x; j < FD; j += 256) {
        g_lds[j] = 1.0f / (1.0f + __expf(-TEMP_ * theta[j]));
        w_lds[j] = weight[j];
    }
    __syncthreads();

    const int lane = threadIdx.x & 31;
    const int h    = lane >> 4;    // half-wave selects K pair {2h, 2h+1}
    const int m    = lane & 15;    // row within the 16-row tile
    const int wave = blockIdx.x * (256 / 32) + (threadIdx.x >> 5);
    const int row_base = wave * 16;
    const int row      = row_base + m;

    v8f acc = {};   // 16x16 f32 C/D accumulator (all 16 columns identical)

    for (int k0 = 0; k0 < FD; k0 += 4) {
        const int f  = k0 >> 4;          // 16 columns per feature
        const int j0 = k0 + 2 * h;       // this lane's two columns
        const int d0 = j0 & 15;          // even -> 8B-aligned float2 load

        // Direct term: one coalesced 8B load from this row's embedding.
        const int idx0 = inp[row * N_FEAT + f];
        const float* erow = emb + ((size_t)f * VOCAB + (size_t)idx0) * EMB_D;
        const v2f xv = *(const v2f*)(erow + d0);

        // Shuffled term: per-column permuted row, element gathers (L2-resident).
        const int  bs0  = perm_row((unsigned)j0,     (unsigned)row);
        const int  bs1  = perm_row((unsigned)(j0+1), (unsigned)row);
        const int  ids0 = inp[bs0 * N_FEAT + f];
        const int  ids1 = inp[bs1 * N_FEAT + f];
        const float xs0 = emb[((size_t)f * VOCAB + (size_t)ids0) * EMB_D + d0];
        const float xs1 = emb[((size_t)f * VOCAB + (size_t)ids1) * EMB_D + d0 + 1];

        const float g0 = g_lds[j0];
        const float g1 = g_lds[j0 + 1];

        v2f a;  // A-matrix 16x4 f32 layout: lanes 0-15 K={0,1}, lanes 16-31 K={2,3}
        a.x = xv.x * g0 + (1.0f - g0) * xs0;
        a.y = xv.y * g1 + (1.0f - g1) * xs1;

        v2f bv; // B = weight broadcast to all 16 columns -> D[m][n] = dot, any n
        bv.x = w_lds[j0];
        bv.y = w_lds[j0 + 1];

        acc = __builtin_amdgcn_wmma_f32_16x16x4_f32(
            /*neg_a=*/false, a, /*neg_b=*/false, bv,
            /*c_mod=*/(short)0, acc, /*reuse_a=*/false, /*reuse_b=*/false);
    }

    // C/D layout: VGPR r holds row (r + 8h), column = lane&15. All columns are
    // equal (B broadcast), so lane 0 has rows 0..7, lane 16 has rows 8..15.
    if (m == 0) {
        const int base = row_base + (h ? 8 : 0);
#pragma unroll
        for (int r = 0; r < 8; ++r) out[base + r] = acc[r];
    }
}

__global__ __launch_bounds__(256) void fs_loss_kernel(
    const float* __restrict__ theta, float* __restrict__ out)
{
    __shared__ float red[256];
    float s = 0.0f;
    for (int j = threadIdx.x; j < FD; j += 256) {
        const int d = j & 15;                      // dim index within feature
        const float g = 1.0f / (1.0f + __expf(-TEMP_ * theta[j]));
        s += g * (0.1f / (float)(d + 1));          // FS_WEIGHT=1, harmonic dim w
    }
    red[threadIdx.x] = s;
    __syncthreads();
    for (int off = 128; off > 0; off >>= 1) {
        if (threadIdx.x < off) red[threadIdx.x] += red[threadIdx.x + off];
        __syncthreads();
    }
    if (threadIdx.x == 0) out[B_ROWS] = red[0] * (1.0f / (float)FD);
}

extern "C" void kernel_launch(void* const* d_in, const int* in_sizes, int n_in,
                              void* d_out, int out_size, void* d_ws, size_t ws_size,
                              hipStream_t stream) {
    const int*   inp    = (const int*)  d_in[0];   // [B, F] indices
    const float* emb    = (const float*)d_in[1];   // [F, V, D]
    const float* theta  = (const float*)d_in[2];   // [F, D]
    const float* weight = (const float*)d_in[3];   // [F, D]
    float* out = (float*)d_out;                    // [B] outputs + [1] fs_loss

    // 256 threads = 8 waves/WG, each wave owns a 16-row strip -> 128 WGs.
    gated_emb_dot_kernel<<<B_ROWS / 128, 256, 0, stream>>>(inp, emb, theta, weight, out);
    fs_loss_kernel<<<1, 256, 0, stream>>>(theta, out);
}